// TripletLoss_81810537055054
// MI455X (gfx1250) — compile-verified
//
#include <hip/hip_runtime.h>

typedef float v2f __attribute__((ext_vector_type(2)));
typedef float v8f __attribute__((ext_vector_type(8)));

#define NROWS 8192
#define DDIM  256
#define MARGINF 0.3f

// ---------------------------------------------------------------------------
// Kernel 1: per-row squared norms + initialize ordered-uint max/min slots.
// One wave32 per row; grid = NROWS/8 blocks of 256 threads.
// ---------------------------------------------------------------------------
__global__ __launch_bounds__(256) void rownorm_init_kernel(
    const float* __restrict__ X, float* __restrict__ sq,
    unsigned int* __restrict__ ap_bits, unsigned int* __restrict__ an_bits)
{
    const int row  = blockIdx.x * 8 + (threadIdx.x >> 5);
    const int lane = threadIdx.x & 31;
    const float* r = X + (size_t)row * DDIM;
    float s = 0.f;
    #pragma unroll
    for (int k = lane; k < DDIM; k += 32) { float v = r[k]; s += v * v; }
    #pragma unroll
    for (int m = 16; m >= 1; m >>= 1) s += __shfl_xor(s, m, 32);
    if (lane == 0) {
        sq[row]      = s;
        ap_bits[row] = 0u;           // identity for max over non-negative floats
        an_bits[row] = 0x7F800000u;  // +inf bits: identity for min over positive floats
    }
}

// ---------------------------------------------------------------------------
// Kernel 2: fused Gram (f32 WMMA 16x16x4) + batch-hard mining.
// Block = 256 threads (8 waves). blockIdx.x -> 16-row i-tile (LDS resident),
// blockIdx.y -> 1/4 chunk of j-tiles; each wave owns one 16-col j-tile per
// iteration and accumulates hardest-pos/neg in registers.
// ---------------------------------------------------------------------------
__global__ __launch_bounds__(256) void mine_kernel(
    const float* __restrict__ X, const int* __restrict__ tgt,
    const float* __restrict__ sq,
    unsigned int* __restrict__ ap_bits, unsigned int* __restrict__ an_bits)
{
    __shared__ float        lA[16 * DDIM];   // 16 KB i-tile
    __shared__ float        lsqi[16];
    __shared__ int          ltgti[16];
    __shared__ unsigned int lap[16], lan[16];

    const int ibase = blockIdx.x * 16;
    const int tid   = threadIdx.x;
    const int wave  = tid >> 5;
    const int lane  = tid & 31;

    { // stage i-tile into LDS (1024 float4 chunks)
        const float4* src = (const float4*)(X + (size_t)ibase * DDIM);
        float4*       dst = (float4*)lA;
        for (int v = tid; v < 16 * DDIM / 4; v += 256) dst[v] = src[v];
    }
    if (tid < 16) {
        lsqi[tid]  = sq[ibase + tid];
        ltgti[tid] = tgt[ibase + tid];
        lap[tid]   = 0u;
        lan[tid]   = 0x7F800000u;
    }
    __syncthreads();

    // f32 16x16x4 fragment addressing: row = lane&15, K-pair selected by lane half
    const int m    = lane & 15;
    const int koff = (lane >> 4) * 2;
    const int half = lane >> 4;

    float runmax[8], runmin[8], msq[8];
    int   mtg[8];
    #pragma unroll
    for (int r = 0; r < 8; ++r) {
        runmax[r] = 0.0f;
        runmin[r] = __builtin_inff();
        msq[r]    = lsqi[r + 8 * half];   // C/D layout: M = r + 8*(lane>>4)
        mtg[r]    = ltgti[r + 8 * half];
    }

    const int jt0 = blockIdx.y * (NROWS / 16 / 4);   // 128 j-tiles per chunk
    for (int it = 0; it < 16; ++it) {                // wave-uniform trip count
        const int jbase = (jt0 + it * 8 + wave) * 16;
        const float* Arow = lA + m * DDIM + koff;                       // LDS
        const float* Brow = X + (size_t)(jbase + m) * DDIM + koff;      // global (L2)
        v8f c = {};
        #pragma unroll 8
        for (int k0 = 0; k0 < DDIM; k0 += 4) {
            v2f a = *(const v2f*)(Arow + k0);   // ds_load_b64
            v2f b = *(const v2f*)(Brow + k0);   // global_load_b64
            c = __builtin_amdgcn_wmma_f32_16x16x4_f32(
                    false, a, false, b, (short)0, c, false, false);
        }
        const float sqj = sq[jbase + m];        // column N == lane&15 == m
        const int   tj  = tgt[jbase + m];
        #pragma unroll
        for (int r = 0; r < 8; ++r) {
            float d2   = msq[r] + sqj - 2.0f * c[r];
            float dist = __builtin_sqrtf(fmaxf(d2, 1e-12f));
            if (mtg[r] == tj) runmax[r] = fmaxf(runmax[r], dist);
            else              runmin[r] = fminf(runmin[r], dist);
        }
    }

    // reduce over the 16 lanes sharing the same M (xor<16 stays in a half)
    #pragma unroll
    for (int r = 0; r < 8; ++r) {
        float mx = runmax[r], mn = runmin[r];
        #pragma unroll
        for (int s = 1; s < 16; s <<= 1) {
            mx = fmaxf(mx, __shfl_xor(mx, s, 32));
            mn = fminf(mn, __shfl_xor(mn, s, 32));
        }
        if ((lane & 15) == 0) {
            const int M = r + 8 * half;
            atomicMax(&lap[M], __float_as_uint(mx));
            atomicMin(&lan[M], __float_as_uint(mn));
        }
    }
    __syncthreads();
    if (tid < 16) {
        atomicMax(&ap_bits[ibase + tid], lap[tid]);
        atomicMin(&an_bits[ibase + tid], lan[tid]);
    }
}

// ---------------------------------------------------------------------------
// Kernel 3: reduce 8192 rows -> (loss, prec)
// ---------------------------------------------------------------------------
__global__ __launch_bounds__(1024) void finalize_kernel(
    const unsigned int* __restrict__ ap_bits,
    const unsigned int* __restrict__ an_bits,
    float* __restrict__ out)
{
    __shared__ float sl[32], sp[32];
    float l = 0.f, p = 0.f;
    for (int i = threadIdx.x; i < NROWS; i += 1024) {
        float a = __uint_as_float(ap_bits[i]);
        float b = __uint_as_float(an_bits[i]);
        l += fmaxf(a - b + MARGINF, 0.0f);
        p += (b > a) ? 1.0f : 0.0f;
    }
    #pragma unroll
    for (int s = 16; s >= 1; s >>= 1) {
        l += __shfl_xor(l, s, 32);
        p += __shfl_xor(p, s, 32);
    }
    const int wv = threadIdx.x >> 5, ln = threadIdx.x & 31;
    if (ln == 0) { sl[wv] = l; sp[wv] = p; }
    __syncthreads();
    if (threadIdx.x < 32) {
        l = sl[threadIdx.x];
        p = sp[threadIdx.x];
        #pragma unroll
        for (int s = 16; s >= 1; s >>= 1) {
            l += __shfl_xor(l, s, 32);
            p += __shfl_xor(p, s, 32);
        }
        if (threadIdx.x == 0) {
            out[0] = l * (1.0f / NROWS);
            out[1] = p * (1.0f / NROWS);
        }
    }
}

extern "C" void kernel_launch(void* const* d_in, const int* in_sizes, int n_in,
                              void* d_out, int out_size, void* d_ws, size_t ws_size,
                              hipStream_t stream) {
    const float* X   = (const float*)d_in[0];
    const int*   tgt = (const int*)d_in[1];

    float*        sq      = (float*)d_ws;
    unsigned int* ap_bits = (unsigned int*)(sq + NROWS);
    unsigned int* an_bits = ap_bits + NROWS;
    float*        out     = (float*)d_out;

    rownorm_init_kernel<<<NROWS / 8, 256, 0, stream>>>(X, sq, ap_bits, an_bits);

    dim3 gridM(NROWS / 16, 4);
    mine_kernel<<<gridM, 256, 0, stream>>>(X, tgt, sq, ap_bits, an_bits);

    finalize_kernel<<<1, 1024, 0, stream>>>(ap_bits, an_bits, out);
}